// MirasMemory_3728031613274
// MI455X (gfx1250) — compile-verified
//
#include <hip/hip_runtime.h>
#include <hip/hip_bf16.h>
#include <math.h>

// Problem constants (from reference setup_inputs)
#define BATCH 16
#define SEQ   4096
#define DK    256
#define DV    256
#define ROWS  16          // rows of M handled per block
#define NT    64          // threads per block (2 waves of 32)

typedef __attribute__((ext_vector_type(2))) float v2f;
typedef __attribute__((ext_vector_type(8))) float v8f;

// LDS byte offset of a __shared__ object (addrspacecast generic -> AS(3))
__device__ __forceinline__ uint32_t lds_addr(const void* p) {
    return (uint32_t)(uintptr_t)(__attribute__((address_space(3))) const void*)p;
}

// Async copy of 16 bytes global -> LDS (GVS mode), tracked by ASYNCcnt.
__device__ __forceinline__ void async_copy16(uint32_t lds_byte, uint32_t vbyte,
                                             const float* sbase) {
    asm volatile("global_load_async_to_lds_b128 %0, %1, %2 offset:0"
                 :
                 : "v"(lds_byte), "v"(vbyte), "s"(sbase)
                 : "memory");
}

// One block owns rows [row0, row0+16) of (M, mom) for one batch, state in LDS.
// Per step: async-prefetched k,q land in LDS; kn = k/(||k||+eps);
// WMMA f32 16x16x4: D(16x16) = M_block(16x256) x X(256x16), X col0=q, col1=kn
// (split-K over the 2 waves, LDS reduction); then rank-1 + momentum update.
__global__ __launch_bounds__(NT) void miras_scan_kernel(
    const float* __restrict__ Kin, const float* __restrict__ Vin,
    const float* __restrict__ Qin, const float* __restrict__ lr_scale,
    float* __restrict__ out, float* __restrict__ Mf, float* __restrict__ momf)
{
    __shared__ __align__(16) float Mst[ROWS * DK];   // 16 KB
    __shared__ __align__(16) float Mom[ROWS * DK];   // 16 KB
    __shared__ __align__(16) float kbuf[2][DK];      // async k landing buffers
    __shared__ __align__(16) float qbuf[2][DK];      // async q landing buffers
    __shared__ __align__(16) float kn[DK];           // normalized key
    __shared__ __align__(16) float zpad[DK];         // zeros for B columns n>=2
    __shared__ float vv[ROWS];
    __shared__ float scr[2 * ROWS * 16];             // split-K partial D tiles
    __shared__ float red2[2];
    __shared__ float scal[2 * ROWS];                 // [0..15]=g, [16..31]=pred

    const int t    = threadIdx.x;
    const int lane = t & 31;
    const int wave = t >> 5;
    const int b    = blockIdx.x >> 4;        // batch
    const int rg   = blockIdx.x & 15;        // row group
    const int row0 = rg * ROWS;

    const float lr = 0.2f / (1.0f + __expf(-lr_scale[0]));  // sigmoid(x)*0.1*2

    // zero-init state + zero pad
    for (int i = t; i < ROWS * DK; i += NT) { Mst[i] = 0.0f; Mom[i] = 0.0f; }
    for (int i = t; i < DK; i += NT) zpad[i] = 0.0f;

    const float* Kb = Kin + (size_t)b * SEQ * DK;
    const float* Qb = Qin + (size_t)b * SEQ * DK;
    const float* Vb = Vin + (size_t)b * SEQ * DV;
    float*       Ob = out + (size_t)b * SEQ * DV + row0;

    const int m  = lane & 15;   // A: row within tile; B: column n
    const int kh = lane >> 4;   // K half within the 4-wide K slab
    const int kbase0 = wave * (DK / 2) + 2 * kh;  // split-K + lane K offset

    const uint32_t vbyte = (uint32_t)t * 16u;     // 16B per thread = 1KB/step

    // prime the pipeline: issue async loads for step 0 into buffer 0
    async_copy16(lds_addr(&kbuf[0][t * 4]), vbyte, Kb);
    async_copy16(lds_addr(&qbuf[0][t * 4]), vbyte, Qb);

    for (int s = 0; s < SEQ; ++s) {
        const int cur = s & 1, nxt = cur ^ 1;

        // issue next step's async loads, then retire the current step's
        if (s + 1 < SEQ) {
            async_copy16(lds_addr(&kbuf[nxt][t * 4]), vbyte, Kb + (size_t)(s + 1) * DK);
            async_copy16(lds_addr(&qbuf[nxt][t * 4]), vbyte, Qb + (size_t)(s + 1) * DK);
            asm volatile("s_wait_asynccnt 0x2" ::: "memory");
        } else {
            asm volatile("s_wait_asynccnt 0x0" ::: "memory");
        }
        __syncthreads();   // all waves' step-s data now visible in LDS

        // ---- ||k||^2 : 4 elems/thread + wave shfl reduction ----
        float ss = 0.0f;
        #pragma unroll
        for (int j = 0; j < 4; ++j) {
            float kv = kbuf[cur][t * 4 + j];
            ss += kv * kv;
        }
        #pragma unroll
        for (int off = 16; off > 0; off >>= 1) ss += __shfl_xor(ss, off, 32);
        if (lane == 0) red2[wave] = ss;
        if (t < ROWS) vv[t] = Vb[(size_t)s * DV + row0 + t];
        __syncthreads();

        const float rn = 1.0f / (__builtin_sqrtf(red2[0] + red2[1]) + 1e-6f);
        #pragma unroll
        for (int j = 0; j < 4; ++j) kn[t * 4 + j] = kbuf[cur][t * 4 + j] * rn;
        __syncthreads();

        // ---- WMMA: acc += A(16x4 of M) x B(4x16 of [q|kn|0..]) over K/2 ----
        const float* arow  = &Mst[m * DK + kbase0];
        const float* bbase = (m == 0) ? &qbuf[cur][kbase0]
                           : (m == 1) ? &kn[kbase0]
                                      : &zpad[kbase0 & (DK - 1)];
        v8f acc = {};
        #pragma unroll 8
        for (int kt = 0; kt < DK / 2; kt += 4) {
            v2f a  = *(const v2f*)(arow + kt);
            v2f bm = *(const v2f*)(bbase + kt);
            acc = __builtin_amdgcn_wmma_f32_16x16x4_f32(
                false, a, false, bm, (short)0, acc, false, false);
        }
        // store partial D tile: element r of lane -> (row r+8*kh, col m)
        #pragma unroll
        for (int r = 0; r < 8; ++r)
            scr[wave * 256 + (r + 8 * kh) * 16 + m] = acc[r];
        __syncthreads();

        // ---- reduce split-K partials; emit output + per-row scalars ----
        if (t < 2 * ROWS) {
            const int mm  = t & 15;
            const int col = t >> 4;           // 0 = q-column (output), 1 = kn (pred)
            float val = scr[mm * 16 + col] + scr[256 + mm * 16 + col];
            if (col == 0) {
                Ob[(size_t)s * DV + mm] = val;
            } else {
                float pred = val;
                scal[mm]        = 2.0f * (pred - vv[mm]);   // g
                scal[ROWS + mm] = pred;
            }
        }
        __syncthreads();

        // ---- state update: mom = .9 mom + g*kn ; M = M - lr*mom - pred*kn ----
        for (int idx = t; idx < ROWS * (DK / 4); idx += NT) {
            const int r  = idx >> 6;              // DK/4 = 64 chunks per row
            const int c4 = (idx & 63) * 4;
            const float g    = scal[r];
            const float pred = scal[ROWS + r];
            float4 kv = *(const float4*)&kn[c4];
            float4 mo = *(float4*)&Mom[r * DK + c4];
            float4 mv = *(float4*)&Mst[r * DK + c4];
            mo.x = 0.9f * mo.x + g * kv.x;  mo.y = 0.9f * mo.y + g * kv.y;
            mo.z = 0.9f * mo.z + g * kv.z;  mo.w = 0.9f * mo.w + g * kv.w;
            mv.x = mv.x - lr * mo.x - pred * kv.x;
            mv.y = mv.y - lr * mo.y - pred * kv.y;
            mv.z = mv.z - lr * mo.z - pred * kv.z;
            mv.w = mv.w - lr * mo.w - pred * kv.w;
            *(float4*)&Mom[r * DK + c4] = mo;
            *(float4*)&Mst[r * DK + c4] = mv;
        }
        __syncthreads();
    }

    // ---- write final M, mom (rows row0..row0+15 contiguous) ----
    float* Mfb  = Mf   + ((size_t)b * DV + row0) * DK;
    float* mofb = momf + ((size_t)b * DV + row0) * DK;
    for (int i = t; i < ROWS * DK; i += NT) {
        Mfb[i]  = Mst[i];
        mofb[i] = Mom[i];
    }
}

extern "C" void kernel_launch(void* const* d_in, const int* in_sizes, int n_in,
                              void* d_out, int out_size, void* d_ws, size_t ws_size,
                              hipStream_t stream) {
    const float* keys     = (const float*)d_in[0];
    const float* values   = (const float*)d_in[1];
    const float* queries  = (const float*)d_in[2];
    const float* lr_scale = (const float*)d_in[3];
    // d_in[4] = retention_scale (unused; its gradient term is identically zero)

    float* out  = (float*)d_out;                               // (B,S,DV)
    float* Mf   = out + (size_t)BATCH * SEQ * DV;              // (B,DV,DK)
    float* momf = Mf  + (size_t)BATCH * DV * DK;               // (B,DV,DK)

    dim3 grid(BATCH * (DV / ROWS));   // 256 blocks
    dim3 block(NT);                   // 64 threads = 2 waves
    miras_scan_kernel<<<grid, block, 0, stream>>>(keys, values, queries, lr_scale,
                                                  out, Mf, momf);
}